// GCN_3788161155718
// MI455X (gfx1250) — compile-verified
//
#include <hip/hip_runtime.h>

typedef float v2f __attribute__((ext_vector_type(2)));
typedef float v8f __attribute__((ext_vector_type(8)));

#define F 128
#define NCLS 16
#define NGRAPH 64

// ---------------- degree / normalization ----------------

__global__ void gcn_deg_init(float* deg, int n) {
  int i = blockIdx.x * blockDim.x + threadIdx.x;
  if (i < n) deg[i] = 1.0f;  // self-loop contribution
}

__global__ void gcn_deg_edge(const int* __restrict__ col, float* deg, int E) {
  int e = blockIdx.x * blockDim.x + threadIdx.x;
  if (e < E) unsafeAtomicAdd(&deg[col[e]], 1.0f);
}

__global__ void gcn_dinv(float* deg, int n) {
  int i = blockIdx.x * blockDim.x + threadIdx.x;
  if (i < n) { float d = deg[i]; deg[i] = d > 0.f ? rsqrtf(d) : 0.f; }
}

__global__ void gcn_norm(const int* __restrict__ row, const int* __restrict__ col,
                         const float* __restrict__ dinv, float* __restrict__ nrm, int E) {
  int e = blockIdx.x * blockDim.x + threadIdx.x;
  if (e < E) nrm[e] = dinv[row[e]] * dinv[col[e]];
}

// ---------------- fp32 WMMA GEMM: out[n x 128] = act(A[n x 128]) @ W[128 x 128] ----------------
// One wave computes a 16(M) x 128(N) strip: 8 accumulators, K-loop of 32 x (8 WMMA).
// W staged in LDS (64 KB). RELU fuses the previous layer's activation into the A-load.

template <bool RELU>
__global__ void __launch_bounds__(256) gcn_gemm(const float* __restrict__ A,
                                                const float* __restrict__ W,
                                                float* __restrict__ out, int n) {
  __shared__ float sW[F * F];
  const int t = threadIdx.x;
  {
    const float4* Wv = (const float4*)W;
    float4* sWv = (float4*)sW;
#pragma unroll
    for (int i = 0; i < 16; ++i) sWv[t + 256 * i] = Wv[t + 256 * i];
  }
  __syncthreads();

  const int wave = t >> 5;          // wave32
  const int lane = t & 31;
  const int rowBase = blockIdx.x * 128 + wave * 16;
  if (rowBase >= n) return;         // wave-uniform: EXEC stays all-1s for WMMA

  const int half = lane >> 4;       // K-pair selector for 16x4 f32 A fragment
  const int lm = lane & 15;
  int m = rowBase + lm;
  if (m >= n) m = n - 1;            // clamp; clamped rows are never stored
  const float* Arow = A + (size_t)m * F;

  v8f acc[8];
#pragma unroll
  for (int nt = 0; nt < 8; ++nt)
#pragma unroll
    for (int v = 0; v < 8; ++v) acc[nt][v] = 0.0f;

  for (int k0 = 0; k0 < F; k0 += 4) {
    const int kb = k0 + 2 * half;
    v2f a;
    a.x = Arow[kb];
    a.y = Arow[kb + 1];
    if (RELU) { a.x = fmaxf(a.x, 0.f); a.y = fmaxf(a.y, 0.f); }
#pragma unroll
    for (int nt = 0; nt < 8; ++nt) {
      v2f b;  // B 4x16 f32 fragment: lane half selects K pair, lm selects N
      b.x = sW[kb * F + nt * 16 + lm];
      b.y = sW[(kb + 1) * F + nt * 16 + lm];
      acc[nt] = __builtin_amdgcn_wmma_f32_16x16x4_f32(
          /*neg_a=*/false, a, /*neg_b=*/false, b,
          /*c_mod=*/(short)0, acc[nt], /*reuse_a=*/false, /*reuse_b=*/false);
    }
  }

  // C/D layout: VGPR v -> M = v + 8*half, N = lm
#pragma unroll
  for (int nt = 0; nt < 8; ++nt) {
#pragma unroll
    for (int v = 0; v < 8; ++v) {
      const int mo = rowBase + v + 8 * half;
      if (mo < n) out[(size_t)mo * F + nt * 16 + lm] = acc[nt][v];
    }
  }
}

// ---------------- aggregation ----------------

// out[i,:] = bias + dinv[i]^2 * h[i,:]   (self-loop term, non-atomic init)
__global__ void gcn_selfbias(const float* __restrict__ h, const float* __restrict__ dinv,
                             const float* __restrict__ bias, float* __restrict__ out, int n) {
  int idx = blockIdx.x * blockDim.x + threadIdx.x;
  if (idx >= n * F) return;
  int i = idx >> 7, f = idx & (F - 1);
  float di = dinv[i];
  out[idx] = bias[f] + di * di * h[idx];
}

// out[col[e],:] += norm[e] * h[row[e],:]   (32 lanes per edge, float4 each)
__global__ void gcn_edge_msg(const int* __restrict__ row, const int* __restrict__ col,
                             const float* __restrict__ nrm, const float* __restrict__ h,
                             float* __restrict__ out, int E) {
  int idx = blockIdx.x * blockDim.x + threadIdx.x;
  int e = idx >> 5;
  if (e >= E) return;
  int f4 = idx & 31;
  float w = nrm[e];
  const float4 v = ((const float4*)(h + (size_t)row[e] * F))[f4];
  float* o = out + (size_t)col[e] * F + f4 * 4;
  unsafeAtomicAdd(o + 0, w * v.x);
  unsafeAtomicAdd(o + 1, w * v.y);
  unsafeAtomicAdd(o + 2, w * v.z);
  unsafeAtomicAdd(o + 3, w * v.w);
}

// ---------------- pooling + head ----------------

__global__ void gcn_pool_zero(float* sums, float* cnt) {
  int idx = blockIdx.x * blockDim.x + threadIdx.x;
  if (idx < NGRAPH * F) sums[idx] = 0.f;
  if (idx < NGRAPH) cnt[idx] = 0.f;
}

__global__ void gcn_pool_acc(const float* __restrict__ h, const int* __restrict__ batch,
                             float* sums, float* cnt, int n) {
  int idx = blockIdx.x * blockDim.x + threadIdx.x;
  if (idx >= n * F) return;
  int i = idx >> 7, f = idx & (F - 1);
  int g = batch[i];
  unsafeAtomicAdd(&sums[g * F + f], h[idx]);
  if (f == 0) unsafeAtomicAdd(&cnt[g], 1.0f);
}

__global__ void gcn_head(const float* __restrict__ sums, const float* __restrict__ cnt,
                         const float* __restrict__ linW, const float* __restrict__ linb,
                         float* __restrict__ out) {
  __shared__ float pooled[F];
  __shared__ float sl[NCLS];
  __shared__ float red[2];
  int g = blockIdx.x, t = threadIdx.x;
  float c = fmaxf(cnt[g], 1.0f);
  pooled[t] = sums[g * F + t] / c;
  __syncthreads();
  if (t < NCLS) {
    float acc = linb[t];
    for (int f = 0; f < F; ++f) acc += pooled[f] * linW[f * NCLS + t];
    sl[t] = acc;
  }
  __syncthreads();
  if (t == 0) { float mx = sl[0]; for (int i = 1; i < NCLS; ++i) mx = fmaxf(mx, sl[i]); red[0] = mx; }
  __syncthreads();
  if (t < NCLS) sl[t] = __expf(sl[t] - red[0]);
  __syncthreads();
  if (t == 0) { float s = 0.f; for (int i = 0; i < NCLS; ++i) s += sl[i]; red[1] = s; }
  __syncthreads();
  if (t < NCLS) out[g * NCLS + t] = sl[t] / red[1];
}

// ---------------- launch ----------------

extern "C" void kernel_launch(void* const* d_in, const int* in_sizes, int n_in,
                              void* d_out, int out_size, void* d_ws, size_t ws_size,
                              hipStream_t stream) {
  const float* x    = (const float*)d_in[0];
  const int*   ei   = (const int*)d_in[1];
  const int*   batch= (const int*)d_in[2];
  const float* W0   = (const float*)d_in[3];
  const float* b0   = (const float*)d_in[4];
  const float* W1   = (const float*)d_in[5];
  const float* b1   = (const float*)d_in[6];
  const float* W2   = (const float*)d_in[7];
  const float* b2   = (const float*)d_in[8];
  const float* linW = (const float*)d_in[9];
  const float* linb = (const float*)d_in[10];
  float* out = (float*)d_out;

  const int n = in_sizes[0] / F;   // 50000
  const int E = in_sizes[1] / 2;   // 800000
  const int* row = ei;             // edge_index[0]
  const int* col = ei + E;         // edge_index[1]

  // workspace: dinv[n] | norm[E] | hA[n*F] | hB[n*F] | psum[64*F] | pcnt[64]  (~54.7 MB)
  float* ws   = (float*)d_ws;
  float* dinv = ws;
  float* nrm  = dinv + n;
  float* hA   = nrm + E;
  float* hB   = hA + (size_t)n * F;
  float* psum = hB + (size_t)n * F;
  float* pcnt = psum + NGRAPH * F;

  const int BT = 256;
  dim3 blk(BT);
  int gN    = (n + BT - 1) / BT;
  int gE    = (E + BT - 1) / BT;
  int gNF   = (n * F + BT - 1) / BT;
  int gE32  = (E * 32 + BT - 1) / BT;
  int gGemm = (n + 127) / 128;

  gcn_deg_init<<<gN, blk, 0, stream>>>(dinv, n);
  gcn_deg_edge<<<gE, blk, 0, stream>>>(col, dinv, E);
  gcn_dinv    <<<gN, blk, 0, stream>>>(dinv, n);
  gcn_norm    <<<gE, blk, 0, stream>>>(row, col, dinv, nrm, E);

  // layer 0
  gcn_gemm<false><<<gGemm, blk, 0, stream>>>(x, W0, hA, n);
  gcn_selfbias   <<<gNF, blk, 0, stream>>>(hA, dinv, b0, hB, n);
  gcn_edge_msg   <<<gE32, blk, 0, stream>>>(row, col, nrm, hA, hB, E);
  // layer 1 (ReLU of layer-0 output fused into A-load)
  gcn_gemm<true> <<<gGemm, blk, 0, stream>>>(hB, W1, hA, n);
  gcn_selfbias   <<<gNF, blk, 0, stream>>>(hA, dinv, b1, hB, n);
  gcn_edge_msg   <<<gE32, blk, 0, stream>>>(row, col, nrm, hA, hB, E);
  // layer 2
  gcn_gemm<true> <<<gGemm, blk, 0, stream>>>(hB, W2, hA, n);
  gcn_selfbias   <<<gNF, blk, 0, stream>>>(hA, dinv, b2, hB, n);
  gcn_edge_msg   <<<gE32, blk, 0, stream>>>(row, col, nrm, hA, hB, E);

  gcn_pool_zero<<<(NGRAPH * F + BT - 1) / BT, blk, 0, stream>>>(psum, pcnt);
  gcn_pool_acc <<<gNF, blk, 0, stream>>>(hB, batch, psum, pcnt, n);
  gcn_head     <<<NGRAPH, dim3(F), 0, stream>>>(psum, pcnt, linW, linb, out);
}